// _GATEncoder_16587163697666
// MI455X (gfx1250) — compile-verified
//
#include <hip/hip_runtime.h>
#include <math.h>

// Problem constants (match reference)
#define NN      50000
#define EE      800000
#define ETOT    (EE + NN)        // edges + self loops = 850000
#define IN_DIM  256
#define HID     64
#define HEADS   4
#define C1      (HID * HEADS)    // 256
#define BN_EPS  1e-5f
#define NEG_SLOPE 0.2f

typedef __attribute__((ext_vector_type(2)))  float  v2f;
typedef __attribute__((ext_vector_type(8)))  float  v8f;

// ---------------------------------------------------------------------------
// fp32 WMMA GEMM: C[M,Nc] = A[M,K] * B[K,Nc].
// Block = 5 waves (160 threads). Each block: 5 row-tiles x 1 col-tile.
// B tile (K x 16) staged in LDS once per block, shared by all 5 waves.
// K, Nc compile-time -> fully unrolled K loop, immediate addressing,
// 64 back-to-back v_wmma_f32_16x16x4_f32 per wave.
// Requires: grid.x = M/80, grid.y = Nc/16, K % 4 == 0.
// ---------------------------------------------------------------------------
template <int K, int Nc>
__global__ __launch_bounds__(160) void gemm_wmma(const float* __restrict__ A,
                                                 const float* __restrict__ B,
                                                 float* __restrict__ C) {
  __shared__ float lds_b[K * 16];
  const int tid  = threadIdx.x;
  const int lane = tid & 31;
  const int wave = tid >> 5;                 // 0..4
  const int col0 = blockIdx.y << 4;

  // Cooperative load of B tile [K][16] into LDS (row-major, 16 floats/row).
  for (int i = tid; i < K * 16; i += 160) {
    int kk = i >> 4, cc = i & 15;
    lds_b[i] = B[(size_t)kk * Nc + col0 + cc];
  }
  __syncthreads();

  const int row0 = (blockIdx.x * 5 + wave) << 4;
  const int r  = lane & 15;                  // M (A) / N (B,C) within tile
  const int g  = lane >> 4;                  // lane group 0/1
  const int kb = g << 1;                     // K sub-offset {0,2}

  // A: 16x4 fp32 tile per wmma; lane holds A[row0+r][k+kb .. k+kb+1]
  const float* __restrict__ Ap = A + (size_t)(row0 + r) * K + kb;
  // B (LDS): lane holds B[k+kb][col0+r], B[k+kb+1][col0+r]
  const float* __restrict__ Bl = lds_b + (kb << 4) + r;

  v8f acc = {};
#pragma unroll
  for (int k = 0; k < K; k += 4) {
    v2f a, b;
    a.x = Ap[k];                 // global_load_b64, immediate offset
    a.y = Ap[k + 1];
    b.x = Bl[k << 4];            // ds loads, immediate offsets (64B apart)
    b.y = Bl[(k + 1) << 4];
    acc = __builtin_amdgcn_wmma_f32_16x16x4_f32(false, a, false, b,
                                                (short)0, acc, false, false);
  }

  // C/D layout: VGPR i: lanes0-15 -> M=i, lanes16-31 -> M=8+i; N = lane&15.
  const int mrow = row0 + (g << 3);
  float* __restrict__ Cp = C + (size_t)mrow * Nc + col0 + r;
#pragma unroll
  for (int i = 0; i < 8; ++i) Cp[(size_t)i * Nc] = acc[i];
}

// ---------------------------------------------------------------------------
// Per-node attention coefficients: as[n,h] = <h[n,h,:], a_src[h,:]>, same dst.
// ---------------------------------------------------------------------------
__global__ void node_alpha(const float* __restrict__ h,
                           const float* __restrict__ a_s,
                           const float* __restrict__ a_d,
                           float* __restrict__ as_out,
                           float* __restrict__ ad_out,
                           int H, int D) {
  int idx = blockIdx.x * blockDim.x + threadIdx.x;
  if (idx >= NN * H) return;
  int n = idx / H, hh = idx - n * H;
  const float4* hp  = (const float4*)(h + (size_t)n * H * D + (size_t)hh * D);
  const float4* asp = (const float4*)(a_s + (size_t)hh * D);
  const float4* adp = (const float4*)(a_d + (size_t)hh * D);
  float s1 = 0.f, s2 = 0.f;
  for (int d = 0; d < D / 4; ++d) {
    float4 v = hp[d], va = asp[d], vb = adp[d];
    s1 += v.x * va.x + v.y * va.y + v.z * va.z + v.w * va.w;
    s2 += v.x * vb.x + v.y * vb.y + v.z * vb.z + v.w * vb.w;
  }
  as_out[idx] = s1;
  ad_out[idx] = s2;
}

__global__ void fill_kernel(float* __restrict__ p, long long n, float v) {
  long long i = (long long)blockIdx.x * blockDim.x + threadIdx.x;
  if (i < n) p[i] = v;
}

__device__ __forceinline__ void atomicMaxF(float* addr, float val) {
  unsigned int* ia = (unsigned int*)addr;
  unsigned int old = __hip_atomic_load(ia, __ATOMIC_RELAXED, __HIP_MEMORY_SCOPE_AGENT);
  while (__uint_as_float(old) < val) {
    unsigned int assumed = old;
    old = atomicCAS(ia, assumed, __float_as_uint(val));
    if (old == assumed) break;
  }
}

__device__ __forceinline__ void edge_sd(const long long* __restrict__ ei,
                                        int e, int* s, int* d) {
  if (e < EE) {
    *s = (int)ei[e];
    *d = (int)ei[EE + e];
  } else {
    *s = *d = e - EE;    // self loop
  }
}

// Pass 1: e = leaky_relu(asrc[src]+adst[dst]); segment max over dst.
__global__ void edge_logits_max(const long long* __restrict__ ei,
                                const float* __restrict__ asrc,
                                const float* __restrict__ adst,
                                float* __restrict__ ebuf,
                                float* __restrict__ mbuf, int H) {
  long long idx = (long long)blockIdx.x * blockDim.x + threadIdx.x;
  if (idx >= (long long)ETOT * H) return;
  int e = (int)(idx / H), h = (int)(idx - (long long)e * H);
  int s, d;
  edge_sd(ei, e, &s, &d);
  float v = asrc[(size_t)s * H + h] + adst[(size_t)d * H + h];
  v = (v > 0.f) ? v : NEG_SLOPE * v;
  ebuf[idx] = v;
  atomicMaxF(&mbuf[(size_t)d * H + h], v);
}

// Pass 2: t = exp(e - m[dst]); segment sum over dst. ebuf overwritten with t.
__global__ void edge_exp_sum(const long long* __restrict__ ei,
                             const float* __restrict__ mbuf,
                             float* __restrict__ ebuf,
                             float* __restrict__ sbuf, int H) {
  long long idx = (long long)blockIdx.x * blockDim.x + threadIdx.x;
  if (idx >= (long long)ETOT * H) return;
  int e = (int)(idx / H), h = (int)(idx - (long long)e * H);
  int s, d;
  edge_sd(ei, e, &s, &d);
  float t = __expf(ebuf[idx] - mbuf[(size_t)d * H + h]);
  ebuf[idx] = t;
  atomicAdd(&sbuf[(size_t)d * H + h], t);
}

// Pass 3 (dominant): agg[dst, c:c+4] += h[src, c:c+4] * t / (s[dst]+eps).
// One thread per (edge, 4-channel group): float4 gather + 4 atomics.
__global__ void edge_aggregate(const long long* __restrict__ ei,
                               const float* __restrict__ hfeat,
                               const float* __restrict__ ebuf,
                               const float* __restrict__ sbuf,
                               float* __restrict__ agg, int H, int C) {
  const int C4 = C >> 2;
  long long w = (long long)blockIdx.x * blockDim.x + threadIdx.x;
  if (w >= (long long)ETOT * C4) return;
  int e = (int)(w / C4), q = (int)(w - (long long)e * C4);
  int c = q << 2;
  int h = c / (C / H);                    // D = C/H divisible by 4
  int s, d;
  edge_sd(ei, e, &s, &d);
  float t   = ebuf[(long long)e * H + h];
  float den = sbuf[(size_t)d * H + h] + 1e-16f;
  float alpha = t / den;
  const float4 hv = *(const float4*)(hfeat + (size_t)s * C + c);
  float* ap = agg + (size_t)d * C + c;
  atomicAdd(ap + 0, hv.x * alpha);
  atomicAdd(ap + 1, hv.y * alpha);
  atomicAdd(ap + 2, hv.z * alpha);
  atomicAdd(ap + 3, hv.w * alpha);
}

// Fused: y = BN(x + bias); out = ELU(y)
__global__ void bias_bn_elu(const float* __restrict__ in,
                            const float* __restrict__ bias,
                            const float* __restrict__ gamma,
                            const float* __restrict__ beta,
                            const float* __restrict__ mean,
                            const float* __restrict__ var,
                            float* __restrict__ out, int C) {
  long long idx = (long long)blockIdx.x * blockDim.x + threadIdx.x;
  if (idx >= (long long)NN * C) return;
  int c = (int)(idx % C);
  float y = in[idx] + bias[c];
  y = (y - mean[c]) * (gamma[c] * rsqrtf(var[c] + BN_EPS)) + beta[c];
  out[idx] = (y > 0.f) ? y : expm1f(y);
}

static inline unsigned int nblk(long long n, int b) {
  return (unsigned int)((n + b - 1) / b);
}

extern "C" void kernel_launch(void* const* d_in, const int* in_sizes, int n_in,
                              void* d_out, int out_size, void* d_ws, size_t ws_size,
                              hipStream_t stream) {
  const float*     x      = (const float*)d_in[0];
  const long long* ei     = (const long long*)d_in[1];   // int64 [2, E]
  const float*     W1     = (const float*)d_in[2];
  const float*     a_src1 = (const float*)d_in[3];
  const float*     a_dst1 = (const float*)d_in[4];
  const float*     b1     = (const float*)d_in[5];
  const float*     bn1g   = (const float*)d_in[6];
  const float*     bn1b   = (const float*)d_in[7];
  const float*     bn1m   = (const float*)d_in[8];
  const float*     bn1v   = (const float*)d_in[9];
  const float*     W2     = (const float*)d_in[10];
  const float*     a_src2 = (const float*)d_in[11];
  const float*     a_dst2 = (const float*)d_in[12];
  const float*     b2     = (const float*)d_in[13];
  const float*     bn2g   = (const float*)d_in[14];
  const float*     bn2b   = (const float*)d_in[15];
  const float*     bn2m   = (const float*)d_in[16];
  const float*     bn2v   = (const float*)d_in[17];
  float* out = (float*)d_out;

  // Workspace layout (floats)
  float* ws = (float*)d_ws;
  size_t off = 0;
  float* h1   = ws + off; off += (size_t)NN * C1;       // x @ W1           [N,256]
  float* agg1 = ws + off; off += (size_t)NN * C1;       // layer1 agg / h2  [N,256]
  float* hw2  = ws + off; off += (size_t)NN * HID;      // h2 @ W2          [N,64]
  float* agg2 = ws + off; off += (size_t)NN * HID;      // layer2 agg       [N,64]
  float* as1  = ws + off; off += (size_t)NN * HEADS;
  float* ad1  = ws + off; off += (size_t)NN * HEADS;
  float* m1   = ws + off; off += (size_t)NN * HEADS;
  float* s1   = ws + off; off += (size_t)NN * HEADS;
  float* e1   = ws + off; off += (size_t)ETOT * HEADS;
  float* as2  = ws + off; off += (size_t)NN;
  float* ad2  = ws + off; off += (size_t)NN;
  float* m2   = ws + off; off += (size_t)NN;
  float* s2   = ws + off; off += (size_t)NN;
  float* e2   = ws + off; off += (size_t)ETOT;
  (void)ws_size; (void)in_sizes; (void)n_in; (void)out_size;

  const int B = 256;

  // ---------------- Layer 1 (H=4, D=64, C=256) ----------------
  {
    dim3 grid(NN / 80, C1 / 16);   // 625 x 16, 5 row-tiles per block
    gemm_wmma<IN_DIM, C1><<<grid, 160, 0, stream>>>(x, W1, h1);
  }
  node_alpha<<<nblk((long long)NN * HEADS, B), B, 0, stream>>>(
      h1, a_src1, a_dst1, as1, ad1, HEADS, HID);
  fill_kernel<<<nblk((long long)NN * HEADS, B), B, 0, stream>>>(
      m1, (long long)NN * HEADS, -INFINITY);
  fill_kernel<<<nblk((long long)NN * HEADS, B), B, 0, stream>>>(
      s1, (long long)NN * HEADS, 0.f);
  fill_kernel<<<nblk((long long)NN * C1, B), B, 0, stream>>>(
      agg1, (long long)NN * C1, 0.f);
  edge_logits_max<<<nblk((long long)ETOT * HEADS, B), B, 0, stream>>>(
      ei, as1, ad1, e1, m1, HEADS);
  edge_exp_sum<<<nblk((long long)ETOT * HEADS, B), B, 0, stream>>>(
      ei, m1, e1, s1, HEADS);
  edge_aggregate<<<nblk((long long)ETOT * (C1 / 4), B), B, 0, stream>>>(
      ei, h1, e1, s1, agg1, HEADS, C1);
  // h2 = ELU(BN(agg1 + b1)), in place over agg1
  bias_bn_elu<<<nblk((long long)NN * C1, B), B, 0, stream>>>(
      agg1, b1, bn1g, bn1b, bn1m, bn1v, agg1, C1);

  // ---------------- Layer 2 (H=1, D=64, C=64) ----------------
  {
    dim3 grid(NN / 80, HID / 16);  // 625 x 4
    gemm_wmma<C1, HID><<<grid, 160, 0, stream>>>(agg1, W2, hw2);
  }
  node_alpha<<<nblk((long long)NN, B), B, 0, stream>>>(
      hw2, a_src2, a_dst2, as2, ad2, 1, HID);
  fill_kernel<<<nblk((long long)NN, B), B, 0, stream>>>(
      m2, (long long)NN, -INFINITY);
  fill_kernel<<<nblk((long long)NN, B), B, 0, stream>>>(
      s2, (long long)NN, 0.f);
  fill_kernel<<<nblk((long long)NN * HID, B), B, 0, stream>>>(
      agg2, (long long)NN * HID, 0.f);
  edge_logits_max<<<nblk((long long)ETOT, B), B, 0, stream>>>(
      ei, as2, ad2, e2, m2, 1);
  edge_exp_sum<<<nblk((long long)ETOT, B), B, 0, stream>>>(
      ei, m2, e2, s2, 1);
  edge_aggregate<<<nblk((long long)ETOT * (HID / 4), B), B, 0, stream>>>(
      ei, hw2, e2, s2, agg2, 1, HID);
  bias_bn_elu<<<nblk((long long)NN * HID, B), B, 0, stream>>>(
      agg2, b2, bn2g, bn2b, bn2m, bn2v, out, HID);
}